// RiemannianGeometry_31181462569459
// MI455X (gfx1250) — compile-verified
//
#include <hip/hip_runtime.h>
#include <hip/hip_bf16.h>
#include <math.h>
#include <stdint.h>

typedef float v2f __attribute__((ext_vector_type(2)));
typedef float v8f __attribute__((ext_vector_type(8)));

#define EPS_FD 0.001f
#define P_DIM 16
#define D_DIM 64

// ---------------------------------------------------------------------------
// Kernel 1: M[p] = clip(tril(L[p]),-5,5) @ clip(tril(L[p]),-5,5)^T + 1e-5*I
// One wave per p. Uses V_WMMA_F32_16X16X4_F32 (f32 WMMA, K=4 per step).
// For this symmetric Gram product the A(16x4) and B(4x16) per-lane layouts
// coincide:  lane%16 = M-row (A) / N-col (B);  VGPR0 = K(2*half), VGPR1 = K+1.
// ---------------------------------------------------------------------------
__device__ __forceinline__ float tril_clamp(const float* __restrict__ L, int r, int d) {
  float v = L[r * D_DIM + d];
  v = fminf(fmaxf(v, -5.0f), 5.0f);
  return (d <= r) ? v : 0.0f;
}

__global__ void chol_gram_wmma_kernel(const float* __restrict__ chol,
                                      float* __restrict__ Mout) {
  const int p    = blockIdx.x;        // 16 blocks of 1 wave
  const int lane = (int)threadIdx.x;  // 0..31
  const int half = lane >> 4;
  const int l16  = lane & 15;
  const float* Lp = chol + p * D_DIM * D_DIM;
  float*       Mp = Mout + p * D_DIM * D_DIM;

  for (int ii = 0; ii < 4; ++ii) {
    for (int jj = 0; jj < 4; ++jj) {
      v8f c = {0.f, 0.f, 0.f, 0.f, 0.f, 0.f, 0.f, 0.f};
      const int rowA = ii * 16 + l16;
      const int rowB = jj * 16 + l16;
#pragma unroll
      for (int kk = 0; kk < 16; ++kk) {           // K = 64 in steps of 4
        const int d0 = kk * 4 + half * 2;
        v2f a, b;
        a.x = tril_clamp(Lp, rowA, d0);
        a.y = tril_clamp(Lp, rowA, d0 + 1);
        b.x = tril_clamp(Lp, rowB, d0);
        b.y = tril_clamp(Lp, rowB, d0 + 1);
        c = __builtin_amdgcn_wmma_f32_16x16x4_f32(false, a, false, b,
                                                  (short)0, c, false, false);
      }
#pragma unroll
      for (int r = 0; r < 8; ++r) {               // C/D layout: row = r + 8*half
        const int row = ii * 16 + r + half * 8;
        const int col = jj * 16 + l16;
        float v = c[r];
        if (row == col) v += 1e-5f;
        Mp[row * D_DIM + col] = v;
      }
    }
  }
}

// ---------------------------------------------------------------------------
// Kernel 2: bandwidth-bound streaming pass over positions (128 MiB).
// Async global->LDS (CDNA5 ASYNCcnt path) into a padded layout, then
// thread-per-row compute from LDS, half-wave shfl reduction over P=16.
// ---------------------------------------------------------------------------
__device__ __forceinline__ void async_copy_b128(unsigned lds_addr, const void* gaddr) {
  asm volatile("global_load_async_to_lds_b128 %0, %1, off"
               :: "v"(lds_addr), "v"(gaddr)
               : "memory");
}

__device__ __forceinline__ float dcf_dev(float sq, float xi) {
  float base = sq - xi * xi;
  float np = sqrtf(base + (xi + EPS_FD) * (xi + EPS_FD));
  float nm = sqrtf(base + (xi - EPS_FD) * (xi - EPS_FD));
  return (tanhf(np) - tanhf(nm)) * (0.1f / (2.0f * EPS_FD));
}

__global__ __launch_bounds__(256) void curvature_kernel(
    const float* __restrict__ pos, const float* __restrict__ Mws,
    const int* __restrict__ kidx, const int* __restrict__ lidx,
    float* __restrict__ out, int n_tiles) {
  // 8 waves/block; each wave owns 2 tiles (= 2 (b,t) pairs) of 16 rows x 64 f32.
  // Row stride in LDS is 68 floats (272 B): 16 B pad -> conflict-free
  // thread-per-row b128 reads (bank = (4*lane + 4*j) % 64).
  __shared__ float lds[8 * 32 * 68];   // 69632 B

  const int lane = (int)(threadIdx.x & 31u);
  const int wave = (int)(threadIdx.x >> 5u);
  const int half = lane >> 4;
  const int l16  = lane & 15;
  const int tile0 = ((int)blockIdx.x * 8 + wave) * 2;
  if (tile0 >= n_tiles) return;

  // ---- async copy: 8 KB contiguous global -> padded LDS rows ----
  const char* gbase   = (const char*)pos + (size_t)tile0 * 4096;  // 2 tiles
  unsigned lds_base   = (unsigned)(uintptr_t)&lds[wave * 32 * 68];
  unsigned lds_lane   = lds_base + (unsigned)(half * 272 + l16 * 16);
  const char* g_lane  = gbase + lane * 16;
#pragma unroll
  for (int i = 0; i < 16; ++i) {  // iter i moves rows 2i,2i+1 (512 B / wave)
    async_copy_b128(lds_lane + (unsigned)(i * 544), g_lane + (size_t)i * 512);
  }
  asm volatile("s_wait_asynccnt 0" ::: "memory");

  // ---- per-lane: one full p-row (lane<16 -> tile0, lane>=16 -> tile0+1) ----
  const float* rowf = &lds[(wave * 32 + lane) * 68];
  const float4* rv = (const float4*)rowf;
  float sq = 0.f;
#pragma unroll
  for (int j = 0; j < 16; ++j) {
    float4 v = rv[j];
    sq += v.x * v.x + v.y * v.y + v.z * v.z + v.w * v.w;
  }

  float acck[4], accl[4];
#pragma unroll
  for (int s = 0; s < 4; ++s) {
    const int k = kidx[s];
    const int l = lidx[s];
    const float mkl = Mws[l16 * (D_DIM * D_DIM) + k * D_DIM + l];
    const float mlk = Mws[l16 * (D_DIM * D_DIM) + l * D_DIM + k];
    const float gk = dcf_dev(sq, rowf[k]) * mkl;
    const float gl = dcf_dev(sq, rowf[l]) * mlk;
    acck[s] = gk * gk;
    accl[s] = gl * gl;
  }

  // ---- sum over P=16 rows = 16 lanes of this half-wave ----
#pragma unroll
  for (int m = 1; m < 16; m <<= 1) {
#pragma unroll
    for (int s = 0; s < 4; ++s) {
      acck[s] += __shfl_xor(acck[s], m, 32);
      accl[s] += __shfl_xor(accl[s], m, 32);
    }
  }

  if (l16 == 0) {
    float r = 0.f;
#pragma unroll
    for (int s = 0; s < 4; ++s) r += sqrtf(acck[s]) + sqrtf(accl[s]);
    out[tile0 + half] = -2.5f * r;   // -10 * mean over S=4
  }
}

// ---------------------------------------------------------------------------
extern "C" void kernel_launch(void* const* d_in, const int* in_sizes, int n_in,
                              void* d_out, int out_size, void* d_ws, size_t ws_size,
                              hipStream_t stream) {
  const float* positions = (const float*)d_in[0];  // (B,T,P,D) f32
  const float* chol      = (const float*)d_in[1];  // (P,D,D)   f32
  const int*   kidx      = (const int*)d_in[2];    // (S,)
  const int*   lidx      = (const int*)d_in[3];    // (S,)
  float*       out       = (float*)d_out;          // (B,T)
  float*       Mws       = (float*)d_ws;           // 16*64*64 f32 = 256 KB

  chol_gram_wmma_kernel<<<P_DIM, 32, 0, stream>>>(chol, Mws);

  const int n_tiles = in_sizes[0] / (P_DIM * D_DIM);  // B*T = 32768
  const int blocks  = (n_tiles + 15) / 16;            // 16 tiles per block
  curvature_kernel<<<blocks, 256, 0, stream>>>(positions, Mws, kidx, lidx, out, n_tiles);
}